// Pseudo_Loss_68341519614543
// MI455X (gfx1250) — compile-verified
//
#include <hip/hip_runtime.h>

namespace {
constexpr int   kN        = 16384;
constexpr int   kD        = 512;
constexpr int   kK        = 512;
constexpr float kTemp     = 0.1f;
constexpr float kEps      = 1e-8f;
constexpr int   kMaxIters = 100;
constexpr float kRtol     = 1e-4f;
constexpr float kAtol     = 1e-8f;
// TDM pads 2 DWORDs after every 256 DWORDs stored -> row pitch 512+4 floats.
// 32 lanes x ds_load_b64 at pitch 516 hit all 64 banks exactly once.
constexpr int   kPitch    = 516;
constexpr int   kALds     = 64 * kPitch * 4;              // A panel: 64 rows
constexpr int   kCLds     = 16 * kPitch * 4;              // one C panel: 16 rows
constexpr int   kTileLds  = kALds + 2 * kCLds;            // + double-buffered C
}

typedef __attribute__((ext_vector_type(2))) float        v2f;
typedef __attribute__((ext_vector_type(8))) float        v8f;
typedef __attribute__((ext_vector_type(4))) unsigned int u32x4;
typedef __attribute__((ext_vector_type(8))) int          i32x8;
typedef __attribute__((ext_vector_type(4))) int          i32x4;

// ---------------------------------------------------------------------------
// TDM: DMA a [rows x 512] f32 row-major panel (row stride 512) from global to
// LDS, inserting 2 pad DWORDs after every 256 DWORDs (-> LDS pitch 516).
__device__ __forceinline__ void tdm_load_panel(const float* gsrc, void* ldsDst, int rows) {
  unsigned long long ga = (unsigned long long)(uintptr_t)gsrc;
  unsigned ldsOff = (unsigned)(uintptr_t)ldsDst;          // LDS aperture: low 32 bits = byte offset
  u32x4 g0;
  g0[0] = 1u;                                             // count=1, user mode, no gather
  g0[1] = ldsOff;                                         // lds_addr
  g0[2] = (unsigned)(ga & 0xFFFFFFFFu);                   // global_addr[31:0]
  g0[3] = (unsigned)((ga >> 32) & 0x01FFFFFFu)            // global_addr[56:32]
        | (2u << 30);                                     // type = 2 (image/tensor)
  i32x8 g1;
  g1[0] = (2 << 16)                                       // data_size = 4 bytes
        | (1 << 20)                                       // pad_enable
        | (7 << 22)                                       // pad_interval: 256 DWORDs
        | (1 << 25);                                      // pad_amount: 2 DWORDs
  g1[1] = (int)((unsigned)(kD & 0xFFFF) << 16);           // tensor_dim0[15:0] = 512
  g1[2] = (int)(((unsigned)kD >> 16) |
                ((unsigned)(rows & 0xFFFF) << 16));       // dim0[31:16] | tensor_dim1[15:0]
  g1[3] = (int)(((unsigned)rows >> 16) |
                ((unsigned)kD << 16));                    // dim1[31:16] | tile_dim0 = 512
  g1[4] = rows;                                           // tile_dim1 = rows, tile_dim2 = 0
  g1[5] = kD;                                             // tensor_dim0_stride = 512
  g1[6] = 0;
  g1[7] = 0;
  i32x4 z4 = {0, 0, 0, 0};                                // groups 2/3 unused (2D tensor)
  i32x8 z8 = {0, 0, 0, 0, 0, 0, 0, 0};                    // extra group (6-arg builtin)
  __builtin_amdgcn_tensor_load_to_lds(g0, g1, z4, z4, z8, 0);
}

// ---------------------------------------------------------------------------
// init: c = x[init_idx], done = 0
__global__ void __launch_bounds__(256)
gather_kernel(const float* __restrict__ x, const int* __restrict__ init_idx,
              float* __restrict__ c, int* __restrict__ flags) {
  int idx = blockIdx.x * 256 + threadIdx.x;          // grid == K*D/256 exactly
  if (idx == 0) flags[0] = 0;                        // done = 0 (reset every launch)
  int k = idx >> 9;
  int d = idx & (kD - 1);
  c[idx] = x[(size_t)init_idx[k] * kD + d];
}

// per-iteration reset: sums=0, cnt=0, conv=1
__global__ void __launch_bounds__(256)
zero_kernel(float* __restrict__ sums, float* __restrict__ cnt, int* __restrict__ flags) {
  if (flags[0]) return;
  int idx = blockIdx.x * 256 + threadIdx.x;
  sums[idx] = 0.0f;
  if (idx < kK) cnt[idx] = 0.0f;
  if (idx == 0) flags[1] = 1;                        // conv = true until disproven
}

// csq[row] = sum(c[row,:]^2), one wave per row
__global__ void __launch_bounds__(128)
rowsq_kernel(const float* __restrict__ m, float* __restrict__ out,
             const int* __restrict__ flags) {
  if (flags[0]) return;
  int wave = threadIdx.x >> 5, lane = threadIdx.x & 31;
  int row = blockIdx.x * 4 + wave;
  const float* p = m + (size_t)row * kD;
  float s = 0.0f;
  for (int d = lane; d < kD; d += 32) { float v = p[d]; s += v * v; }
#pragma unroll
  for (int o = 16; o > 0; o >>= 1) s += __shfl_xor(s, o, 32);
  if (lane == 0) out[row] = s;
}

// ---------------------------------------------------------------------------
// assignment: ids[i] = argmin_j ( csq[j] - 2 * dot(x_i, c_j) )
// TDM-staged panels + f32 WMMA, double-buffered centroid tiles.
__global__ void __launch_bounds__(128)
assign_kernel(const float* __restrict__ x, const float* __restrict__ cmat,
              const float* __restrict__ csq, int* __restrict__ ids,
              const int* __restrict__ flags) {
  if (flags[0]) return;
  extern __shared__ float lds[];
  float* As  = lds;                         // 64 rows x kPitch
  float* Cs0 = lds + 64 * kPitch;           // 16 rows x kPitch (buffer 0)
  float* Cs1 = Cs0 + 16 * kPitch;           // 16 rows x kPitch (buffer 1)

  const int tid = threadIdx.x;
  const int wave = tid >> 5;
  const int lane = tid & 31;
  const int half = lane >> 4;               // 0: lanes 0-15, 1: lanes 16-31
  const int lm = lane & 15;
  const int rowBase = blockIdx.x * 64;

  if (wave == 0) {
    tdm_load_panel(x + (size_t)rowBase * kD, As, 64);     // A panel, once
    tdm_load_panel(cmat, Cs0, 16);                        // first centroid tile
  }

  float bestVal[8];
  int   bestIdx[8];
#pragma unroll
  for (int r = 0; r < 8; ++r) { bestVal[r] = 3.4e38f; bestIdx[r] = 0; }

  // A frag: lane -> row lm of this wave's slab, k-offset 2*half (16x4 f32 layout)
  // B frag: lane -> centroid row lm of the tile, k-offset 2*half (4x16 layout)
  const float* Arow = As + (wave * 16 + lm) * kPitch + 2 * half;
  const int    bOff = lm * kPitch + 2 * half;

  for (int jt = 0; jt < kK / 16; ++jt) {
    if (wave == 0) __builtin_amdgcn_s_wait_tensorcnt(0);  // DMA for this tile done
    __syncthreads();                                      // publish LDS to all waves
    if (wave == 0 && jt + 1 < kK / 16)                    // prefetch next tile into
      tdm_load_panel(cmat + (size_t)(jt + 1) * 16 * kD,   // the other buffer
                     ((jt + 1) & 1) ? Cs1 : Cs0, 16);

    const float* Brow = ((jt & 1) ? Cs1 : Cs0) + bOff;
    v8f acc = {0.f, 0.f, 0.f, 0.f, 0.f, 0.f, 0.f, 0.f};
#pragma unroll 8
    for (int k = 0; k < 256; k += 4) {                    // first half-row
      v2f a = *reinterpret_cast<const v2f*>(Arow + k);
      v2f b = *reinterpret_cast<const v2f*>(Brow + k);
      acc = __builtin_amdgcn_wmma_f32_16x16x4_f32(false, a, false, b, (short)0, acc,
                                                  false, false);
    }
#pragma unroll 8
    for (int k = 256; k < 512; k += 4) {                  // after mid-row 2-DWORD pad
      v2f a = *reinterpret_cast<const v2f*>(Arow + k + 2);
      v2f b = *reinterpret_cast<const v2f*>(Brow + k + 2);
      acc = __builtin_amdgcn_wmma_f32_16x16x4_f32(false, a, false, b, (short)0, acc,
                                                  false, false);
    }

    const int col = jt * 16 + lm;
    const float cq = csq[col];
#pragma unroll
    for (int r = 0; r < 8; ++r) {
      float sc = cq - 2.0f * acc[r];                      // x_sq[i] constant per row: dropped
      if (sc < bestVal[r]) { bestVal[r] = sc; bestIdx[r] = col; }
    }
  }

  // reduce over the 16 lanes of each half (C/D layout: vgpr r <-> rows r, r+8)
#pragma unroll
  for (int r = 0; r < 8; ++r) {
    float v = bestVal[r];
    int idx = bestIdx[r];
#pragma unroll
    for (int m = 1; m < 16; m <<= 1) {
      float ov = __shfl_xor(v, m, 32);
      int   oi = __shfl_xor(idx, m, 32);
      if (ov < v || (ov == v && oi < idx)) { v = ov; idx = oi; }
    }
    if (lm == 0) ids[rowBase + wave * 16 + r + 8 * half] = idx;
  }
}

// ---------------------------------------------------------------------------
// segment sums via f32 atomics; one wave per row
__global__ void __launch_bounds__(128)
accumulate_kernel(const float* __restrict__ x, const int* __restrict__ ids,
                  float* __restrict__ sums, float* __restrict__ cnt,
                  const int* __restrict__ flags) {
  if (flags[0]) return;
  int wave = threadIdx.x >> 5, lane = threadIdx.x & 31;
  int row = blockIdx.x * 4 + wave;
  int k = ids[row];
  const float* xr = x + (size_t)row * kD;
  float* sr = sums + (size_t)k * kD;
  for (int d = lane; d < kD; d += 32) atomicAdd(&sr[d], xr[d]);
  if (lane == 0) atomicAdd(&cnt[k], 1.0f);
}

// cnew = sums / max(cnt,1); allclose test -> conv flag
__global__ void __launch_bounds__(256)
finalize_kernel(const float* __restrict__ sums, const float* __restrict__ cnt,
                const float* __restrict__ c, float* __restrict__ cnew,
                int* __restrict__ flags) {
  if (flags[0]) return;
  int idx = blockIdx.x * 256 + threadIdx.x;
  int k = idx >> 9;
  float nv = sums[idx] / fmaxf(cnt[k], 1.0f);
  cnew[idx] = nv;
  if (fabsf(c[idx] - nv) > kAtol + kRtol * fabsf(nv)) flags[1] = 0;
}

// conv ? freeze (done=1, torch breaks before updating) : c = cnew
__global__ void __launch_bounds__(256)
update_kernel(const float* __restrict__ cnew, float* __restrict__ c,
              int* __restrict__ flags) {
  if (flags[0]) return;
  int idx = blockIdx.x * 256 + threadIdx.x;
  if (flags[1]) { if (idx == 0) flags[0] = 1; return; }
  c[idx] = cnew[idx];
}

// ---------------------------------------------------------------------------
// cn[row,:] = cnew[row,:] / (||cnew[row,:]|| + eps)
__global__ void __launch_bounds__(128)
cnorm_kernel(const float* __restrict__ cnew, float* __restrict__ cn) {
  int wave = threadIdx.x >> 5, lane = threadIdx.x & 31;
  int row = blockIdx.x * 4 + wave;
  const float* p = cnew + (size_t)row * kD;
  float s = 0.0f;
  for (int d = lane; d < kD; d += 32) { float v = p[d]; s += v * v; }
#pragma unroll
  for (int o = 16; o > 0; o >>= 1) s += __shfl_xor(s, o, 32);
  float scale = 1.0f / (sqrtf(s) + kEps);
  float* q = cn + (size_t)row * kD;
  for (int d = lane; d < kD; d += 32) q[d] = p[d] * scale;
}

// xscale[i] = 1 / ((||x_i|| + eps) * (temp + eps))
__global__ void __launch_bounds__(128)
xscale_kernel(const float* __restrict__ x, float* __restrict__ xscale) {
  int wave = threadIdx.x >> 5, lane = threadIdx.x & 31;
  int row = blockIdx.x * 4 + wave;
  const float* p = x + (size_t)row * kD;
  float s = 0.0f;
  for (int d = lane; d < kD; d += 32) { float v = p[d]; s += v * v; }
#pragma unroll
  for (int o = 16; o > 0; o >>= 1) s += __shfl_xor(s, o, 32);
  if (lane == 0) xscale[row] = 1.0f / ((sqrtf(s) + kEps) * (kTemp + kEps));
}

// ---------------------------------------------------------------------------
// loss: logits = (x @ cn^T) * xscale[i]; online logsumexp + target pick
__global__ void __launch_bounds__(128)
loss_kernel(const float* __restrict__ x, const float* __restrict__ cn,
            const float* __restrict__ xscale, const int* __restrict__ ids,
            float* __restrict__ partials) {
  extern __shared__ float lds[];
  float* As  = lds;
  float* Cs0 = lds + 64 * kPitch;
  float* Cs1 = Cs0 + 16 * kPitch;
  __shared__ float waveSum[4];

  const int tid = threadIdx.x;
  const int wave = tid >> 5;
  const int lane = tid & 31;
  const int half = lane >> 4;
  const int lm = lane & 15;
  const int rowBase = blockIdx.x * 64;

  if (wave == 0) {
    tdm_load_panel(x + (size_t)rowBase * kD, As, 64);
    tdm_load_panel(cn, Cs0, 16);
  }

  int   rid[8];
  float xs[8], mx[8], sm[8], tg[8];
#pragma unroll
  for (int r = 0; r < 8; ++r) {
    int grow = rowBase + wave * 16 + r + 8 * half;
    rid[r] = ids[grow];
    xs[r] = xscale[grow];
    mx[r] = -3.4e38f; sm[r] = 0.0f; tg[r] = -3.4e38f;
  }

  const float* Arow = As + (wave * 16 + lm) * kPitch + 2 * half;
  const int    bOff = lm * kPitch + 2 * half;

  for (int jt = 0; jt < kK / 16; ++jt) {
    if (wave == 0) __builtin_amdgcn_s_wait_tensorcnt(0);
    __syncthreads();
    if (wave == 0 && jt + 1 < kK / 16)
      tdm_load_panel(cn + (size_t)(jt + 1) * 16 * kD, ((jt + 1) & 1) ? Cs1 : Cs0, 16);

    const float* Brow = ((jt & 1) ? Cs1 : Cs0) + bOff;
    v8f acc = {0.f, 0.f, 0.f, 0.f, 0.f, 0.f, 0.f, 0.f};
#pragma unroll 8
    for (int k = 0; k < 256; k += 4) {
      v2f a = *reinterpret_cast<const v2f*>(Arow + k);
      v2f b = *reinterpret_cast<const v2f*>(Brow + k);
      acc = __builtin_amdgcn_wmma_f32_16x16x4_f32(false, a, false, b, (short)0, acc,
                                                  false, false);
    }
#pragma unroll 8
    for (int k = 256; k < 512; k += 4) {
      v2f a = *reinterpret_cast<const v2f*>(Arow + k + 2);
      v2f b = *reinterpret_cast<const v2f*>(Brow + k + 2);
      acc = __builtin_amdgcn_wmma_f32_16x16x4_f32(false, a, false, b, (short)0, acc,
                                                  false, false);
    }

    const int col = jt * 16 + lm;
#pragma unroll
    for (int r = 0; r < 8; ++r) {
      float v = acc[r] * xs[r];
      if (col == rid[r]) tg[r] = v;
      if (v > mx[r]) { sm[r] = sm[r] * expf(mx[r] - v) + 1.0f; mx[r] = v; }
      else           { sm[r] += expf(v - mx[r]); }
    }
  }

  float laneTotal = 0.0f;
#pragma unroll
  for (int r = 0; r < 8; ++r) {
    float mm = mx[r], ss = sm[r], tt = tg[r];
#pragma unroll
    for (int m = 1; m < 16; m <<= 1) {
      float om = __shfl_xor(mm, m, 32);
      float os = __shfl_xor(ss, m, 32);
      float ot = __shfl_xor(tt, m, 32);
      float M = fmaxf(mm, om);
      ss = ss * expf(mm - M) + os * expf(om - M);
      mm = M;
      tt = fmaxf(tt, ot);
    }
    if (lm == 0) laneTotal += (mm + logf(ss)) - tt;   // -logp[i, ids[i]]
  }
  laneTotal += __shfl_xor(laneTotal, 16, 32);
  if (lane == 0) waveSum[wave] = laneTotal;
  __syncthreads();
  if (tid == 0) partials[blockIdx.x] = waveSum[0] + waveSum[1] + waveSum[2] + waveSum[3];
}

// deterministic final tree reduction over 256 block partials
__global__ void __launch_bounds__(256)
reduce_kernel(const float* __restrict__ partials, float* __restrict__ out) {
  __shared__ float buf[256];
  int t = threadIdx.x;
  buf[t] = partials[t];
  __syncthreads();
  for (int s = 128; s > 0; s >>= 1) {
    if (t < s) buf[t] += buf[t + s];
    __syncthreads();
  }
  if (t == 0) out[0] = buf[0] * (1.0f / (float)kN);
}

// ---------------------------------------------------------------------------
extern "C" void kernel_launch(void* const* d_in, const int* in_sizes, int n_in,
                              void* d_out, int out_size, void* d_ws, size_t ws_size,
                              hipStream_t stream) {
  (void)in_sizes; (void)n_in; (void)out_size; (void)ws_size;
  const float* x = (const float*)d_in[0];
  const int* init_idx = (const int*)d_in[1];
  float* out = (float*)d_out;

  char* w = (char*)d_ws;
  auto alloc = [&](size_t bytes) -> char* {
    char* p = w; w += (bytes + 255) & ~(size_t)255; return p;
  };
  float* c        = (float*)alloc((size_t)kK * kD * 4);  // centroids / later cn
  float* cnew     = (float*)alloc((size_t)kK * kD * 4);  // new centroids == final centers
  float* sums     = (float*)alloc((size_t)kK * kD * 4);
  float* cnt      = (float*)alloc((size_t)kK * 4);
  float* csq      = (float*)alloc((size_t)kK * 4);
  float* xscale   = (float*)alloc((size_t)kN * 4);
  float* partials = (float*)alloc(256 * 4);
  int*   ids      = (int*)alloc((size_t)kN * 4);
  int*   flags    = (int*)alloc(2 * 4);                  // [0]=done, [1]=conv

  (void)hipFuncSetAttribute((const void*)assign_kernel,
                            hipFuncAttributeMaxDynamicSharedMemorySize, kTileLds);
  (void)hipFuncSetAttribute((const void*)loss_kernel,
                            hipFuncAttributeMaxDynamicSharedMemorySize, kTileLds);

  gather_kernel<<<kK * kD / 256, 256, 0, stream>>>(x, init_idx, c, flags);

  for (int it = 0; it < kMaxIters; ++it) {
    zero_kernel<<<kK * kD / 256, 256, 0, stream>>>(sums, cnt, flags);
    rowsq_kernel<<<kK / 4, 128, 0, stream>>>(c, csq, flags);
    assign_kernel<<<kN / 64, 128, kTileLds, stream>>>(x, c, csq, ids, flags);
    accumulate_kernel<<<kN / 4, 128, 0, stream>>>(x, ids, sums, cnt, flags);
    finalize_kernel<<<kK * kD / 256, 256, 0, stream>>>(sums, cnt, c, cnew, flags);
    update_kernel<<<kK * kD / 256, 256, 0, stream>>>(cnew, c, flags);
  }

  cnorm_kernel<<<kK / 4, 128, 0, stream>>>(cnew, c);        // c := normalized centers
  xscale_kernel<<<kN / 4, 128, 0, stream>>>(x, xscale);
  loss_kernel<<<kN / 64, 128, kTileLds, stream>>>(x, c, xscale, ids, partials);
  reduce_kernel<<<1, 256, 0, stream>>>(partials, out);
}